// UserVectorGNN_17815524344480
// MI455X (gfx1250) — compile-verified
//
#include <hip/hip_runtime.h>
#include <hip/hip_bf16.h>

typedef __attribute__((ext_vector_type(2))) float v2f;
typedef __attribute__((ext_vector_type(8))) float v8f;

#define N_NODES 50000
#define N_EDGES 800000
#define M_TILES 3125   // 50000 / 16, exact

// ---------------------------------------------------------------------------
// Edge / degree preprocessing
// ---------------------------------------------------------------------------
__global__ void init_deg(float* deg) {
    int i = blockIdx.x * blockDim.x + threadIdx.x;
    if (i < N_NODES) deg[i] = 1.0f;  // self-loop
}

__global__ void prep_edges(const long long* __restrict__ ei,
                           int* __restrict__ src32, int* __restrict__ dst32,
                           float* __restrict__ deg) {
    int e = blockIdx.x * blockDim.x + threadIdx.x;
    if (e >= N_EDGES) return;
    int s = (int)ei[e];
    int t = (int)ei[N_EDGES + e];
    src32[e] = s;
    dst32[e] = t;
    atomicAdd(&deg[t], 1.0f);
}

__global__ void compute_dinv(const float* __restrict__ deg, float* __restrict__ dinv) {
    int i = blockIdx.x * blockDim.x + threadIdx.x;
    if (i < N_NODES) {
        float d = deg[i];
        dinv[i] = (d > 0.0f) ? rsqrtf(d) : 0.0f;
    }
}

__global__ void edge_norm(const int* __restrict__ src32, const int* __restrict__ dst32,
                          const float* __restrict__ dinv, float* __restrict__ norm) {
    int e = blockIdx.x * blockDim.x + threadIdx.x;
    if (e < N_EDGES) norm[e] = dinv[src32[e]] * dinv[dst32[e]];
}

// ---------------------------------------------------------------------------
// WMMA fp32 GEMM: out[M,N] = X[M,K] @ W[K,N]  (+ optional bias+relu epilogue)
// One wave per 16x16 C tile; K stepped by 4 via V_WMMA_F32_16X16X4_F32.
// M = 50000 (3125 tiles), K,N multiples of 16.
// ---------------------------------------------------------------------------
__global__ void gemm16_wmma(const float* __restrict__ X, const float* __restrict__ W,
                            const float* __restrict__ bias, float* __restrict__ out,
                            int K, int N, int fuse_bias_relu) {
    const int mt   = blockIdx.x;
    const int nt   = blockIdx.y;
    const int lane = threadIdx.x;        // 0..31, one wave per block
    const int half = lane >> 4;          // 0: lanes 0-15, 1: lanes 16-31
    const int l15  = lane & 15;
    const int koff = half << 1;          // A/B VGPR0 holds K=0 (lo half) / K=2 (hi half)

    v8f c = {};
    const float* xrow = X + (size_t)(mt * 16 + l15) * K;
    const int    col  = nt * 16 + l15;

    for (int k = 0; k < K; k += 4) {
        v2f a, b;
        // A 16x4 layout (ISA 7.12.2): lane l15 = row M, VGPR0/1 = K={koff,koff+1}
        a.x = xrow[k + koff];
        a.y = xrow[k + koff + 1];
        // B 4x16 layout: row K striped across lanes (N = l15), halves hold K+2
        b.x = W[(size_t)(k + koff) * N + col];
        b.y = W[(size_t)(k + koff + 1) * N + col];
        c = __builtin_amdgcn_wmma_f32_16x16x4_f32(
            /*neg_a=*/false, a, /*neg_b=*/false, b,
            /*c_mod=*/(short)0, c, /*reuse_a=*/false, /*reuse_b=*/false);
    }

    float bv = fuse_bias_relu ? bias[col] : 0.0f;
#pragma unroll
    for (int v = 0; v < 8; ++v) {
        // C/D layout: VGPR v holds M=v (lanes 0-15) / M=v+8 (lanes 16-31)
        int row = mt * 16 + v + (half << 3);
        float val = c[v] + bv;
        if (fuse_bias_relu) val = fmaxf(val, 0.0f);
        out[(size_t)row * N + col] = val;
    }
}

// ---------------------------------------------------------------------------
// Aggregation kernels
// ---------------------------------------------------------------------------
// P[i,f] = H[i,f] * dinv[i]^2   (self-loop contribution; full overwrite of P)
__global__ void scale_init(const float* __restrict__ H, const float* __restrict__ dinv,
                           float* __restrict__ P, int total, int dlog) {
    int tid = blockIdx.x * blockDim.x + threadIdx.x;
    if (tid >= total) return;
    float di = dinv[tid >> dlog];
    P[tid] = H[tid] * di * di;
}

// P[dst,f4..f4+3] += H[src,f4..f4+3] * norm[e]; one thread per (edge, float4)
__global__ void scatter_edges(const float* __restrict__ H,
                              const int* __restrict__ src32, const int* __restrict__ dst32,
                              const float* __restrict__ norm, float* __restrict__ P,
                              int dlog4, int d) {
    int tid = blockIdx.x * blockDim.x + threadIdx.x;
    int e = tid >> dlog4;
    if (e >= N_EDGES) return;
    int f4 = (tid & ((1 << dlog4) - 1)) << 2;
    int s = src32[e], t = dst32[e];
    float nv = norm[e];
    const float4 h = *(const float4*)(H + (size_t)s * d + f4);
    float* p = P + (size_t)t * d + f4;
    atomicAdd(p + 0, h.x * nv);
    atomicAdd(p + 1, h.y * nv);
    atomicAdd(p + 2, h.z * nv);
    atomicAdd(p + 3, h.w * nv);
}

// P[i,f] = relu(P[i,f] + b[f])   (in place)
__global__ void bias_relu(float* __restrict__ P, const float* __restrict__ bias,
                          int total, int dmask) {
    int tid = blockIdx.x * blockDim.x + threadIdx.x;
    if (tid >= total) return;
    P[tid] = fmaxf(P[tid] + bias[tid & dmask], 0.0f);
}

// ---------------------------------------------------------------------------
// Host orchestration
// ---------------------------------------------------------------------------
static inline int cdiv(int a, int b) { return (a + b - 1) / b; }

extern "C" void kernel_launch(void* const* d_in, const int* in_sizes, int n_in,
                              void* d_out, int out_size, void* d_ws, size_t ws_size,
                              hipStream_t stream) {
    const float*     x    = (const float*)d_in[0];
    const long long* ei   = (const long long*)d_in[1];
    const float *W1 = (const float*)d_in[2],  *b1  = (const float*)d_in[3];
    const float *W2 = (const float*)d_in[4],  *b2  = (const float*)d_in[5];
    const float *W3 = (const float*)d_in[6],  *b3  = (const float*)d_in[7];
    const float *Wl1= (const float*)d_in[8],  *bl1 = (const float*)d_in[9];
    const float *Wl2= (const float*)d_in[10], *bl2 = (const float*)d_in[11];
    const float *Wl3= (const float*)d_in[12], *bl3 = (const float*)d_in[13];
    float* out = (float*)d_out;

    // Workspace carving (all buffers fully written before read, every call)
    float* deg   = (float*)d_ws;                  // 50048
    float* dinv  = deg + 50048;                   // 50048
    int*   src32 = (int*)(dinv + 50048);          // 800000
    int*   dst32 = src32 + N_EDGES;               // 800000
    float* norm  = (float*)(dst32 + N_EDGES);     // 800000
    float* bufA  = norm + N_EDGES;                // 50000*128
    float* bufB  = bufA + (size_t)N_NODES * 128;  // 50000*128

    const int TB = 256;

    // ---- Edge preprocessing (shared by all conv layers) ----
    init_deg<<<cdiv(N_NODES, TB), TB, 0, stream>>>(deg);
    prep_edges<<<cdiv(N_EDGES, TB), TB, 0, stream>>>(ei, src32, dst32, deg);
    compute_dinv<<<cdiv(N_NODES, TB), TB, 0, stream>>>(deg, dinv);
    edge_norm<<<cdiv(N_EDGES, TB), TB, 0, stream>>>(src32, dst32, dinv, norm);

    // ---- One GCN conv layer: Xin[K] -> act in P[d] ----
    auto conv = [&](const float* Xin, const float* W, const float* bias,
                    int K, int d, int dlog, float* H, float* P) {
        dim3 g(M_TILES, d / 16);
        gemm16_wmma<<<g, 32, 0, stream>>>(Xin, W, nullptr, H, K, d, 0);
        int total = N_NODES * d;
        scale_init<<<cdiv(total, TB), TB, 0, stream>>>(H, dinv, P, total, dlog);
        int dlog4 = dlog - 2;
        int sthreads = N_EDGES << dlog4;
        scatter_edges<<<cdiv(sthreads, TB), TB, 0, stream>>>(H, src32, dst32, norm, P, dlog4, d);
        bias_relu<<<cdiv(total, TB), TB, 0, stream>>>(P, bias, total, d - 1);
    };

    // conv1: x[16] -> bufB[32]
    conv(x,    W1, b1, 16,  32, 5, bufA, bufB);
    // conv2: bufB[32] -> bufB[64]  (bufB freed after GEMM reads it)
    conv(bufB, W2, b2, 32,  64, 6, bufA, bufB);
    // conv3: bufB[64] -> bufB[128]
    conv(bufB, W3, b3, 64, 128, 7, bufA, bufB);

    // ---- Dense head: fused bias+relu WMMA GEMMs ----
    {   // lin1: 128 -> 64, bufB -> bufA
        dim3 g(M_TILES, 64 / 16);
        gemm16_wmma<<<g, 32, 0, stream>>>(bufB, Wl1, bl1, bufA, 128, 64, 1);
    }
    {   // lin2: 64 -> 32, bufA -> bufB
        dim3 g(M_TILES, 32 / 16);
        gemm16_wmma<<<g, 32, 0, stream>>>(bufA, Wl2, bl2, bufB, 64, 32, 1);
    }
    {   // lin3: 32 -> 16, bufB -> d_out
        dim3 g(M_TILES, 16 / 16);
        gemm16_wmma<<<g, 32, 0, stream>>>(bufB, Wl3, bl3, out, 32, 16, 1);
    }
}